// SMPL_Lres_8074538517084
// MI455X (gfx1250) — compile-verified
//
#include <hip/hip_runtime.h>
#include <hip/hip_bf16.h>

typedef __attribute__((ext_vector_type(2))) float v2f;
typedef __attribute__((ext_vector_type(8))) float v8f;

#define NB   1024
#define NV   6890
#define NC   (NV*3)        // 20670 columns
#define KPD  207
#define KEXT 220           // 207 posedirs + 10 shapedirs(beta) + 1 v_template, pad to mult of 4
#define BT   16            // batch tile
#define VT   16            // vertex tile
#define COLS (VT*3)        // 48

static constexpr int PAR[24] = {-1,0,0,0,1,2,3,4,5,6,7,8,9,9,9,12,13,14,16,17,18,19,20,21};

// ---------------- Kernel 0: SJ[k][j][c] = shapedirs . J_regressor ; baseJ = v_template . J_regressor
__global__ void k0_sj(const float* __restrict__ shapedirs,
                      const float* __restrict__ vtemp,
                      const float* __restrict__ Jreg,
                      float* __restrict__ SJ /* 720 + 72 */) {
  int idx = blockIdx.x * blockDim.x + threadIdx.x;
  if (idx >= 792) return;
  float s = 0.f;
  if (idx < 720) {
    int k = idx / 72, e = idx % 72, j = e / 3, c = e % 3;
    for (int v = 0; v < NV; ++v) s += shapedirs[k * NC + v * 3 + c] * Jreg[v * 24 + j];
  } else {
    int e = idx - 720, j = e / 3, c = e % 3;
    for (int v = 0; v < NV; ++v) s += vtemp[v * 3 + c] * Jreg[v * 24 + j];
  }
  SJ[idx] = s;
}

// ---------------- Kernel 1: Rodrigues -> Rs (B*24*9) and pose_feature (B*207)
__global__ void k1_rod(const float* __restrict__ theta,
                       float* __restrict__ Rs,
                       float* __restrict__ PF) {
  int idx = blockIdx.x * blockDim.x + threadIdx.x;
  if (idx >= NB * 24) return;
  int b = idx / 24, j = idx % 24;
  float t0 = theta[b * 72 + j * 3 + 0];
  float t1 = theta[b * 72 + j * 3 + 1];
  float t2 = theta[b * 72 + j * 3 + 2];
  float a0 = t0 + 1e-8f, a1 = t1 + 1e-8f, a2 = t2 + 1e-8f;
  float ang = sqrtf(a0 * a0 + a1 * a1 + a2 * a2);
  float inv = 1.0f / ang;
  float n0 = t0 * inv, n1 = t1 * inv, n2 = t2 * inv;
  float h = 0.5f * ang;
  float w = cosf(h), sh = sinf(h);
  float x = sh * n0, y = sh * n1, z = sh * n2;
  float qn = rsqrtf(w * w + x * x + y * y + z * z);
  w *= qn; x *= qn; y *= qn; z *= qn;
  float R[9];
  R[0] = w*w + x*x - y*y - z*z; R[1] = 2.f*(x*y - w*z);       R[2] = 2.f*(w*y + x*z);
  R[3] = 2.f*(w*z + x*y);       R[4] = w*w - x*x + y*y - z*z; R[5] = 2.f*(y*z - w*x);
  R[6] = 2.f*(x*z - w*y);       R[7] = 2.f*(w*x + y*z);       R[8] = w*w - x*x - y*y + z*z;
#pragma unroll
  for (int e = 0; e < 9; ++e) Rs[(size_t)idx * 9 + e] = R[e];
  if (j >= 1) {
#pragma unroll
    for (int e = 0; e < 9; ++e)
      PF[(size_t)b * KPD + (j - 1) * 9 + e] = R[e] - ((e % 4 == 0) ? 1.f : 0.f);
  }
}

// ---------------- Kernel 2: per-batch J + kinematic chain -> A (B*24*16)
__global__ void k2_chain(const float* __restrict__ beta,
                         const float* __restrict__ SJ,   // 720 SJ + 72 baseJ
                         const float* __restrict__ Rs,
                         float* __restrict__ A) {
  int b = blockIdx.x * blockDim.x + threadIdx.x;
  if (b >= NB) return;
  const float* bJ = SJ + 720;
  float bet[10];
#pragma unroll
  for (int k = 0; k < 10; ++k) bet[k] = beta[b * 10 + k];
  float J[72];
#pragma unroll
  for (int e = 0; e < 72; ++e) {
    float s = bJ[e];
#pragma unroll
    for (int k = 0; k < 10; ++k) s += bet[k] * SJ[k * 72 + e];
    J[e] = s;
  }
  float res[24][12];  // 3x4 world transforms (bottom row implicit 0001)
  {
    const float* R0 = Rs + (size_t)(b * 24) * 9;
#pragma unroll
    for (int r = 0; r < 3; ++r) {
      res[0][r * 4 + 0] = R0[r * 3 + 0];
      res[0][r * 4 + 1] = R0[r * 3 + 1];
      res[0][r * 4 + 2] = R0[r * 3 + 2];
      res[0][r * 4 + 3] = J[r];
    }
  }
#pragma unroll
  for (int i = 1; i < 24; ++i) {
    const int p = PAR[i];
    const float* Ri = Rs + (size_t)(b * 24 + i) * 9;
    float t0 = J[i * 3 + 0] - J[p * 3 + 0];
    float t1 = J[i * 3 + 1] - J[p * 3 + 1];
    float t2 = J[i * 3 + 2] - J[p * 3 + 2];
#pragma unroll
    for (int r = 0; r < 3; ++r) {
      float p0 = res[p][r * 4 + 0], p1 = res[p][r * 4 + 1], p2 = res[p][r * 4 + 2];
#pragma unroll
      for (int c = 0; c < 3; ++c)
        res[i][r * 4 + c] = p0 * Ri[0 * 3 + c] + p1 * Ri[1 * 3 + c] + p2 * Ri[2 * 3 + c];
      res[i][r * 4 + 3] = p0 * t0 + p1 * t1 + p2 * t2 + res[p][r * 4 + 3];
    }
  }
#pragma unroll
  for (int j = 0; j < 24; ++j) {
    float jx = J[j * 3 + 0], jy = J[j * 3 + 1], jz = J[j * 3 + 2];
    float* Ao = A + (size_t)(b * 24 + j) * 16;
#pragma unroll
    for (int r = 0; r < 3; ++r) {
      float tr = res[j][r * 4 + 0] * jx + res[j][r * 4 + 1] * jy + res[j][r * 4 + 2] * jz;
      Ao[r * 4 + 0] = res[j][r * 4 + 0];
      Ao[r * 4 + 1] = res[j][r * 4 + 1];
      Ao[r * 4 + 2] = res[j][r * 4 + 2];
      Ao[r * 4 + 3] = res[j][r * 4 + 3] - tr;
    }
    Ao[12] = 0.f; Ao[13] = 0.f; Ao[14] = 0.f; Ao[15] = 1.f;
  }
}

// ---------------- Kernel 3: fused v_posed WMMA GEMM + T=weights@A WMMA + skinning
// Block: 128 threads (4 waves); tile = 16 batches x 16 vertices (48 output columns).
// LDS layout (floats):
//   sPF : [0      , 3584 )  16 x 224 extended pose-feature (207 PF + 10 beta + 1 one + pad)
//   sA  : [3584   , 9728 )  16 x 384 joint transforms
//   sW  : [9728   , 10112)  16 x 24  skinning weights
//   sB  : [10112  , 20672)  220 x 48 B-panel (posedirs/shapedirs/v_template)
//   sVP : [10112  , 10880)  16 x 48  v_posed  (reuses sB after barrier)
//   sT  : [10880  , 14976)  16 x 16 x 16 T    (reuses sB after barrier)
#define OFF_PF 0
#define OFF_A  3584
#define OFF_W  9728
#define OFF_B  10112
#define OFF_VP 10112
#define OFF_T  10880

__global__ __launch_bounds__(128)
void k3_fused(const float* __restrict__ beta,
              const float* __restrict__ vtemp,
              const float* __restrict__ shapedirs,
              const float* __restrict__ posedirs,
              const float* __restrict__ weights,
              const float* __restrict__ PF,
              const float* __restrict__ A,
              float* __restrict__ out) {
  __shared__ float smem[20672];
  const int tid  = threadIdx.x;
  const int wave = tid >> 5;
  const int lane = tid & 31;
  const int m    = lane & 15;
  const int kh   = (lane >> 4) << 1;      // 0 or 2
  const int vbase   = blockIdx.x * VT;
  const int colbase = blockIdx.x * COLS;
  const int bbase   = blockIdx.y * BT;

  // ---- cooperative loads ----
  for (int idx = tid; idx < BT * 224; idx += 128) {
    int bb = idx / 224, k = idx % 224;
    float v = 0.f;
    if (k < KPD)       v = PF[(size_t)(bbase + bb) * KPD + k];
    else if (k < 217)  v = beta[(bbase + bb) * 10 + (k - KPD)];
    else if (k == 217) v = 1.0f;
    smem[OFF_PF + idx] = v;
  }
  for (int idx = tid; idx < KEXT * COLS; idx += 128) {
    int k = idx / COLS, c = idx % COLS;
    int col = colbase + c;
    float v = 0.f;
    if (col < NC) {
      if (k < KPD)       v = posedirs[(size_t)k * NC + col];
      else if (k < 217)  v = shapedirs[(size_t)(k - KPD) * NC + col];
      else if (k == 217) v = vtemp[col];
    }
    smem[OFF_B + idx] = v;
  }
  for (int idx = tid; idx < BT * 384; idx += 128) {
    int bb = idx / 384, e = idx % 384;
    smem[OFF_A + idx] = A[(size_t)(bbase + bb) * 384 + e];
  }
  for (int idx = tid; idx < VT * 24; idx += 128) {
    int vv = idx / 24, j = idx % 24;
    int v = vbase + vv;
    smem[OFF_W + idx] = (v < NV) ? weights[(size_t)v * 24 + j] : 0.f;
  }
  __syncthreads();

  // ---- phase 1: v_posed = [PF|beta|1] x [posedirs;shapedirs;v_template]  (waves 0..2) ----
  v8f acc = {};
  if (wave < 3) {
    const int ncb = wave * 16;
    for (int kk = 0; kk < KEXT / 4; ++kk) {
      const int kb = kk * 4;
      v2f a, b;
      a.x = smem[OFF_PF + m * 224 + kb + kh];
      a.y = smem[OFF_PF + m * 224 + kb + kh + 1];
      b.x = smem[OFF_B + (kb + kh) * COLS + ncb + m];
      b.y = smem[OFF_B + (kb + kh + 1) * COLS + ncb + m];
      acc = __builtin_amdgcn_wmma_f32_16x16x4_f32(false, a, false, b,
                                                  (short)0, acc, false, false);
    }
  }
  __syncthreads();   // sB no longer needed; safe to overwrite with sVP/sT
  if (wave < 3) {
    const int ncb = wave * 16;
#pragma unroll
    for (int i = 0; i < 8; ++i) {
      int M = i + ((lane >> 4) << 3);
      smem[OFF_VP + M * COLS + ncb + m] = acc[i];
    }
  }

  // ---- phase 2: T[b] = weights(16x24) @ A[b](24x16) per batch, 4 batches per wave ----
#pragma unroll
  for (int rep = 0; rep < 4; ++rep) {
    const int bb = wave + rep * 4;
    v8f tac = {};
#pragma unroll
    for (int kk = 0; kk < 6; ++kk) {
      const int kb = kk * 4;
      v2f a, b;
      a.x = smem[OFF_W + m * 24 + kb + kh];
      a.y = smem[OFF_W + m * 24 + kb + kh + 1];
      b.x = smem[OFF_A + bb * 384 + (kb + kh) * 16 + m];
      b.y = smem[OFF_A + bb * 384 + (kb + kh + 1) * 16 + m];
      tac = __builtin_amdgcn_wmma_f32_16x16x4_f32(false, a, false, b,
                                                  (short)0, tac, false, false);
    }
#pragma unroll
    for (int i = 0; i < 8; ++i) {
      int M = i + ((lane >> 4) << 3);
      smem[OFF_T + bb * 256 + M * 16 + m] = tac[i];
    }
  }
  __syncthreads();

  // ---- phase 3: skinning, 256 (b,v) pairs over 128 threads ----
  for (int p = tid; p < BT * VT; p += 128) {
    int bb = p >> 4, vv = p & 15;
    int v = vbase + vv;
    if (v >= NV) continue;
    float x = smem[OFF_VP + bb * COLS + vv * 3 + 0];
    float y = smem[OFF_VP + bb * COLS + vv * 3 + 1];
    float z = smem[OFF_VP + bb * COLS + vv * 3 + 2];
    const float* T = &smem[OFF_T + bb * 256 + vv * 16];
    float ox = T[0] * x + T[1] * y + T[2]  * z + T[3];
    float oy = T[4] * x + T[5] * y + T[6]  * z + T[7];
    float oz = T[8] * x + T[9] * y + T[10] * z + T[11];
    size_t o = ((size_t)(bbase + bb) * NV + v) * 3;
    out[o + 0] = ox; out[o + 1] = oy; out[o + 2] = oz;
  }
}

extern "C" void kernel_launch(void* const* d_in, const int* in_sizes, int n_in,
                              void* d_out, int out_size, void* d_ws, size_t ws_size,
                              hipStream_t stream) {
  const float* theta     = (const float*)d_in[0];
  const float* beta      = (const float*)d_in[1];
  const float* vtemp     = (const float*)d_in[2];
  const float* shapedirs = (const float*)d_in[3];
  const float* posedirs  = (const float*)d_in[4];
  const float* Jreg      = (const float*)d_in[5];
  const float* weights   = (const float*)d_in[6];
  float* out = (float*)d_out;

  float* ws = (float*)d_ws;
  float* SJ = ws;                       // 792
  float* Rs = ws + 800;                 // 1024*24*9 = 221184
  float* PF = ws + 800 + 221184;        // 1024*207 = 211968
  float* A  = ws + 800 + 221184 + 211968; // 1024*24*16 = 393216

  k0_sj  <<<dim3(4), dim3(256), 0, stream>>>(shapedirs, vtemp, Jreg, SJ);
  k1_rod <<<dim3((NB * 24 + 255) / 256), dim3(256), 0, stream>>>(theta, Rs, PF);
  k2_chain<<<dim3(NB / 256), dim3(256), 0, stream>>>(beta, SJ, Rs, A);
  dim3 grid((NV + VT - 1) / VT, NB / BT);
  k3_fused<<<grid, dim3(128), 0, stream>>>(beta, vtemp, shapedirs, posedirs,
                                           weights, PF, A, out);
}